// APPNPModel_81209241632807
// MI455X (gfx1250) — compile-verified
//
#include <hip/hip_runtime.h>

// ---------------------------------------------------------------------------
// APPNP GNN for MI455X (gfx1250, wave32).
// GEMMs use V_WMMA_F32_16X16X4_F32 (full f32 precision, matches reference).
// Propagations: self-loop init pass + edge scatter with global f32 atomics;
// h (25.6 MB) fits in the 192 MB L2, so gather/scatter stays on-chip.
// ---------------------------------------------------------------------------

typedef __attribute__((ext_vector_type(2))) float v2f;
typedef __attribute__((ext_vector_type(8))) float v8f;

#define ALPHA 0.1f

// ---------------- degree / normalization ----------------

__global__ void k_deg_init(float* __restrict__ deg, int n) {
    int i = blockIdx.x * blockDim.x + threadIdx.x;
    if (i < n) deg[i] = 1.0f;  // self-loop contributes 1
}

__global__ void k_deg_count(float* __restrict__ deg, const int* __restrict__ dst, int e) {
    int i = blockIdx.x * blockDim.x + threadIdx.x;
    if (i < e) atomicAdd(&deg[dst[i]], 1.0f);
}

__global__ void k_rsqrt(float* __restrict__ d, int n) {
    int i = blockIdx.x * blockDim.x + threadIdx.x;
    if (i < n) {
        float v = d[i];
        d[i] = (v > 0.0f) ? rsqrtf(v) : 0.0f;
    }
}

__global__ void k_edge_w(float* __restrict__ w, const int* __restrict__ src,
                         const int* __restrict__ dst, const float* __restrict__ dinv, int e) {
    int i = blockIdx.x * blockDim.x + threadIdx.x;
    if (i < e) w[i] = dinv[src[i]] * dinv[dst[i]];
}

// ---------------- f32 WMMA GEMM: C[M x NOUT] = A[M x K] @ W[K x NOUT] + b ----------------
// One wave per 16x16 output tile. blockDim.x = 32 * (NOUT/16); gridDim.x = M/16.
// A tile layout (16x4 f32): lane<16 -> row=lane, K=k,k+1 ; lane>=16 -> row=lane-16, K=k+2,k+3
// B tile layout (4x16 f32): lane<16 -> col=lane, K=k,k+1 ; lane>=16 -> col=lane-16, K=k+2,k+3
// C/D: VGPR j: lanes0-15 -> M=j, N=lane ; lanes16-31 -> M=j+8, N=lane-16

template <int K, int NOUT>
__global__ void k_gemm_wmma(const float* __restrict__ A, const float* __restrict__ W,
                            const float* __restrict__ bias, float* __restrict__ C) {
    const int wave = threadIdx.x >> 5;
    const int lane = threadIdx.x & 31;
    const int row0 = blockIdx.x * 16;
    const int col0 = wave * 16;
    const int lidx = lane & 15;          // row (A) / col (B,C)
    const int khalf = (lane >> 4) * 2;   // 0 or 2

    const float* __restrict__ arow = A + (size_t)(row0 + lidx) * K + khalf;
    const float* __restrict__ wcol = W + (size_t)khalf * NOUT + col0 + lidx;

    v8f acc = {};
#pragma unroll 4
    for (int k = 0; k < K; k += 4) {
        v2f a, b;
        a.x = arow[0];
        a.y = arow[1];
        b.x = wcol[0];
        b.y = wcol[NOUT];
        acc = __builtin_amdgcn_wmma_f32_16x16x4_f32(
            /*neg_a=*/false, a, /*neg_b=*/false, b,
            /*c_mod=*/(short)0, acc, /*reuse_a=*/false, /*reuse_b=*/false);
        arow += 4;
        wcol += 4 * NOUT;
    }

    const float bv = bias[col0 + lidx];
    const int moff = (lane >> 4) * 8;
    float* __restrict__ cbase = C + (size_t)row0 * NOUT + col0 + lidx;
#pragma unroll
    for (int j = 0; j < 8; ++j) {
        cbase[(size_t)(j + moff) * NOUT] = acc[j] + bv;
    }
}

// ---------------- propagation ----------------

// out[i,:] = dinv[i]^2 * h[i,:]   (self-loop term; fully initializes out)
template <int F>
__global__ void k_prop_self(float* __restrict__ out, const float* __restrict__ h,
                            const float* __restrict__ dinv, int n) {
    const int vpr = F / 4;
    long long gid = (long long)blockIdx.x * blockDim.x + threadIdx.x;
    long long total = (long long)n * vpr;
    if (gid >= total) return;
    int i = (int)(gid / vpr);
    int f4 = (int)(gid % vpr);
    float sw = dinv[i] * dinv[i];
    float4 hv = ((const float4*)(h + (size_t)i * F))[f4];
    float4 ov = make_float4(sw * hv.x, sw * hv.y, sw * hv.z, sw * hv.w);
    ((float4*)(out + (size_t)i * F))[f4] = ov;
}

// out[dst,:] += w[e] * h[src,:]
template <int F>
__global__ void k_prop_edges(float* __restrict__ out, const float* __restrict__ h,
                             const int* __restrict__ src, const int* __restrict__ dst,
                             const float* __restrict__ w, int e) {
    const int vpr = F / 4;
    long long gid = (long long)blockIdx.x * blockDim.x + threadIdx.x;
    long long total = (long long)e * vpr;
    if (gid >= total) return;
    int ei = (int)(gid / vpr);
    int f4 = (int)(gid % vpr);
    int s = src[ei];
    int d = dst[ei];
    float we = w[ei];
    float4 hv = ((const float4*)(h + (size_t)s * F))[f4];
    float* o = out + (size_t)d * F + (size_t)f4 * 4;
    atomicAdd(o + 0, we * hv.x);
    atomicAdd(o + 1, we * hv.y);
    atomicAdd(o + 2, we * hv.z);
    atomicAdd(o + 3, we * hv.w);
}

// h = (1-alpha)*tmp + alpha*h0
__global__ void k_combine(float* __restrict__ h, const float* __restrict__ tmp,
                          const float* __restrict__ h0, long long n4) {
    long long gid = (long long)blockIdx.x * blockDim.x + threadIdx.x;
    if (gid >= n4) return;
    float4 t = ((const float4*)tmp)[gid];
    float4 z = ((const float4*)h0)[gid];
    float4 r = make_float4((1.0f - ALPHA) * t.x + ALPHA * z.x,
                           (1.0f - ALPHA) * t.y + ALPHA * z.y,
                           (1.0f - ALPHA) * t.z + ALPHA * z.z,
                           (1.0f - ALPHA) * t.w + ALPHA * z.w);
    ((float4*)h)[gid] = r;
}

// ---------------------------------------------------------------------------

static inline int cdiv(long long a, int b) { return (int)((a + b - 1) / b); }

extern "C" void kernel_launch(void* const* d_in, const int* in_sizes, int n_in,
                              void* d_out, int out_size, void* d_ws, size_t ws_size,
                              hipStream_t stream) {
    const float* x  = (const float*)d_in[0];   // [N,256]
    const int*   ei = (const int*)d_in[1];     // [2,E]
    const float* W1 = (const float*)d_in[2];   // [256,128]
    const float* b1 = (const float*)d_in[3];   // [128]
    const float* W3 = (const float*)d_in[4];   // [128,64]
    const float* b3 = (const float*)d_in[5];   // [64]
    float* out = (float*)d_out;                // [N,64]

    const int N = in_sizes[0] / 256;           // 50000
    const int E = in_sizes[1] / 2;             // 800000
    const int* src = ei;
    const int* dst = ei + E;

    // workspace layout (floats, 64-float aligned)
    float* ws = (float*)d_ws;
    float* dinv = ws;                                  // N
    size_t off = ((size_t)N + 63) & ~(size_t)63;
    float* wbuf = ws + off;                            // E
    off += ((size_t)E + 63) & ~(size_t)63;
    float* h0 = ws + off;                              // N*128
    off += (size_t)N * 128;
    float* h = ws + off;                               // N*128
    off += (size_t)N * 128;
    float* tmp = ws + off;                             // N*128
    off += (size_t)N * 128;
    float* hw = ws + off;                              // N*64

    const int T = 256;

    // ---- GCN normalization ----
    k_deg_init<<<cdiv(N, T), T, 0, stream>>>(dinv, N);
    k_deg_count<<<cdiv(E, T), T, 0, stream>>>(dinv, dst, E);
    k_rsqrt<<<cdiv(N, T), T, 0, stream>>>(dinv, N);
    k_edge_w<<<cdiv(E, T), T, 0, stream>>>(wbuf, src, dst, dinv, E);

    // ---- GCNConv 1: tmp = x @ W1 + b1 (WMMA), then h0 = propagate(tmp) ----
    k_gemm_wmma<256, 128><<<N / 16, 32 * (128 / 16), 0, stream>>>(x, W1, b1, tmp);
    k_prop_self<128><<<cdiv((long long)N * 32, T), T, 0, stream>>>(h0, tmp, dinv, N);
    k_prop_edges<128><<<cdiv((long long)E * 32, T), T, 0, stream>>>(h0, tmp, src, dst, wbuf, E);

    // h = h0
    hipMemcpyAsync(h, h0, (size_t)N * 128 * sizeof(float), hipMemcpyDeviceToDevice, stream);

    // ---- APPNP: 10 x { tmp = propagate(h); h = 0.9*tmp + 0.1*h0 } ----
    const long long n4 = (long long)N * 128 / 4;
    for (int it = 0; it < 10; ++it) {
        k_prop_self<128><<<cdiv((long long)N * 32, T), T, 0, stream>>>(tmp, h, dinv, N);
        k_prop_edges<128><<<cdiv((long long)E * 32, T), T, 0, stream>>>(tmp, h, src, dst, wbuf, E);
        k_combine<<<cdiv(n4, T), T, 0, stream>>>(h, tmp, h0, n4);
    }

    // ---- GCNConv 2: hw = h @ W3 + b3 (WMMA), then out = propagate(hw) ----
    k_gemm_wmma<128, 64><<<N / 16, 32 * (64 / 16), 0, stream>>>(h, W3, b3, hw);
    k_prop_self<64><<<cdiv((long long)N * 16, T), T, 0, stream>>>(out, hw, dinv, N);
    k_prop_edges<64><<<cdiv((long long)E * 16, T), T, 0, stream>>>(out, hw, src, dst, wbuf, E);
}